// SenEmbedding_Loss_2353642078406
// MI455X (gfx1250) — compile-verified
//
#include <hip/hip_runtime.h>
#include <cmath>

typedef float v2f __attribute__((ext_vector_type(2)));
typedef float v4f __attribute__((ext_vector_type(4)));
typedef float v8f __attribute__((ext_vector_type(8)));

#define S_  64
#define B_  32
#define D_  768
#define V_  50257
#define NROWS (S_ * B_)        // 2048
#define NPAIR (B_ * (S_ - 1))  // 2016

// ---------------------------------------------------------------------------
// Kernel 1: per-sentence-row Gram statistics via V_WMMA_F32_16X16X4_F32.
// One wave handles (batch b, 16 consecutive sentences s0..s0+15).
// For the f32 16x16x4 shape, the per-lane element mapping of the A operand
// (M=lane&15, K=2*(lane>>4)+j) equals that of the B operand
// (N=lane&15, K=2*(lane>>4)+j), so one float2 load serves as both sides of a
// self-Gram; rows shifted by +1 sentence give consecutive-pair dots on the
// diagonal. Stats per row r=s*B+b: [xx, gg, xg, x_s.x_{s+1}, g_s.g_{s+1}].
// ---------------------------------------------------------------------------
__device__ __forceinline__ float diag16(v8f c, int lane) {
  // C layout: VGPR r holds (M=r, N=lane) for lanes 0-15 and (M=r+8, N=lane-16)
  // for lanes 16-31.  Diag m<8 -> VGPR m, lane m.  Diag m>=8 -> VGPR m-8,
  // lane m+16.  Leave diag[m] in lane m (m = 0..15).
  float d = 0.f;
#pragma unroll
  for (int r = 0; r < 8; ++r) {
    float t1 = __shfl(c[r], r, 32);       // element (r, r)
    float t2 = __shfl(c[r], r + 24, 32);  // element (r+8, r+8)
    if (lane == r)     d = t1;
    if (lane == r + 8) d = t2;
  }
  return d;
}

__global__ __launch_bounds__(32) void gram_kernel(const float* __restrict__ X,
                                                  const float* __restrict__ G,
                                                  float* __restrict__ stats) {
  const int lane = threadIdx.x;
  const int b    = blockIdx.x & (B_ - 1);
  const int s0   = (blockIdx.x >> 5) << 4;
  const int m    = lane & 15;
  const int hi   = lane >> 4;
  const int s    = s0 + m;
  const int s1   = (s + 1 < S_) ? (s + 1) : (S_ - 1);  // clamp: row 63 result unused

  const float* xr  = X + ((size_t)s  * B_ + b) * D_;
  const float* xr1 = X + ((size_t)s1 * B_ + b) * D_;
  const float* gr  = G + ((size_t)s  * B_ + b) * D_;
  const float* gr1 = G + ((size_t)s1 * B_ + b) * D_;

  v8f cxx = {}, cxn = {}, cxy = {}, cgg = {}, cgn = {};

  for (int k = 0; k < D_; k += 4) {
    const int kk = k + 2 * hi;
    v2f ax  = *(const v2f*)(xr  + kk);
    v2f ax1 = *(const v2f*)(xr1 + kk);
    v2f ag  = *(const v2f*)(gr  + kk);
    v2f ag1 = *(const v2f*)(gr1 + kk);
    cxx = __builtin_amdgcn_wmma_f32_16x16x4_f32(false, ax, false, ax,  (short)0, cxx, false, false);
    cxn = __builtin_amdgcn_wmma_f32_16x16x4_f32(false, ax, false, ax1, (short)0, cxn, false, false);
    cxy = __builtin_amdgcn_wmma_f32_16x16x4_f32(false, ax, false, ag,  (short)0, cxy, false, false);
    cgg = __builtin_amdgcn_wmma_f32_16x16x4_f32(false, ag, false, ag,  (short)0, cgg, false, false);
    cgn = __builtin_amdgcn_wmma_f32_16x16x4_f32(false, ag, false, ag1, (short)0, cgn, false, false);
  }

  float dxx = diag16(cxx, lane);
  float dxn = diag16(cxn, lane);
  float dxy = diag16(cxy, lane);
  float dgg = diag16(cgg, lane);
  float dgn = diag16(cgn, lane);

  if (lane < 16) {
    const int row = (s0 + lane) * B_ + b;
    float* st = stats + (size_t)row * 8;
    st[0] = dxx;  // ||x||^2
    st[1] = dgg;  // ||g||^2
    st[2] = dxy;  // x.g
    st[3] = dxn;  // x_s . x_{s+1}
    st[4] = dgn;  // g_s . g_{s+1}
  }
}

// ---------------------------------------------------------------------------
// Kernel 2: cross-entropy.  One block per row (t = b*S + s), single-pass
// online softmax over V=50257.  B128 non-temporal streaming loads (rows are
// only 4B-aligned, so a <=3-element prologue reaches 16B alignment).
// ---------------------------------------------------------------------------
__device__ __forceinline__ void online_update(float v, float& m, float& sum) {
  if (v > m) { sum = sum * __expf(m - v) + 1.f; m = v; }
  else       { sum += __expf(v - m); }
}

__global__ __launch_bounds__(256) void ce_kernel(const float* __restrict__ logits,
                                                 const int* __restrict__ labels,
                                                 float* __restrict__ nll) {
  __shared__ float sm[256];
  __shared__ float ss[256];
  const int t   = blockIdx.x;
  const int tid = threadIdx.x;
  const float* p = logits + (size_t)t * V_;

  // alignment prologue: bring pointer to 16B boundary
  const int mis = (int)(((uintptr_t)p >> 2) & 3);
  const int a   = (4 - mis) & 3;          // 0..3 scalar elements up front
  const int nvec = (V_ - a) >> 2;         // full float4 chunks
  const int tail0 = a + (nvec << 2);
  const int rem  = V_ - tail0;            // 0..3 scalar tail elements

  float m = -INFINITY, sum = 0.f;
  if (tid < a) online_update(p[tid], m, sum);

  const v4f* pv = (const v4f*)(p + a);
  for (int i = tid; i < nvec; i += 256) {
    v4f v = __builtin_nontemporal_load(pv + i);
    online_update(v.x, m, sum);
    online_update(v.y, m, sum);
    online_update(v.z, m, sum);
    online_update(v.w, m, sum);
  }
  if (tid < rem) online_update(p[tail0 + tid], m, sum);

  sm[tid] = m; ss[tid] = sum;
  __syncthreads();
  for (int off = 128; off > 0; off >>= 1) {
    if (tid < off) {
      float m1 = sm[tid], m2 = sm[tid + off];
      float s1 = ss[tid], s2 = ss[tid + off];
      float M = fmaxf(m1, m2);
      ss[tid] = s1 * expf(m1 - M) + s2 * expf(m2 - M);
      sm[tid] = M;
    }
    __syncthreads();
  }
  if (tid == 0) {
    int lab = labels[t];
    float lse = logf(ss[0]) + sm[0];
    nll[t] = (lab != -100) ? (lse - p[lab < 0 ? 0 : lab]) : 0.f;
  }
}

// ---------------------------------------------------------------------------
// Kernel 3: single-block epilogue assembling all five losses.
// ---------------------------------------------------------------------------
__device__ __forceinline__ float blockReduceSum(float v, float* buf) {
  const int tid = threadIdx.x;
  buf[tid] = v;
  __syncthreads();
  for (int off = 128; off > 0; off >>= 1) {
    if (tid < off) buf[tid] += buf[tid + off];
    __syncthreads();
  }
  float r = buf[0];
  __syncthreads();
  return r;
}

__global__ __launch_bounds__(256) void final_kernel(const float* __restrict__ stats,
                                                    const float* __restrict__ nll,
                                                    const int* __restrict__ labels,
                                                    float* __restrict__ out) {
  __shared__ float s_do[NPAIR];
  __shared__ float s_dt[NPAIR];
  __shared__ float s_pm[NPAIR];
  __shared__ float s_red[256];
  const int tid = threadIdx.x;

  // Per-row terms (index idx = s*B + b for stats; any order is fine for sums).
  float vmSum = 0.f, sqSum = 0.f, cosSum = 0.f, ceSum = 0.f, keepSum = 0.f;
  for (int idx = tid; idx < NROWS; idx += 256) {
    const int s = idx >> 5, b = idx & 31;
    const int lab = labels[b * S_ + s];
    const float vm = (lab != 0 && lab != 1) ? 1.f : 0.f;
    const float* st = stats + (size_t)idx * 8;
    const float xx = st[0], yy = st[1], xy = st[2];
    vmSum += vm;
    sqSum += vm * (xx - 2.f * xy + yy);
    const float nx = fmaxf(sqrtf(xx), 1e-8f);
    const float ny = fmaxf(sqrtf(yy), 1e-8f);
    cosSum += vm * (1.f - xy / (nx * ny));
    // CE partial sums (full sum over all rows; index mapping irrelevant)
    ceSum   += nll[idx];
    keepSum += (labels[idx] != -100) ? 1.f : 0.f;
  }

  // Consecutive-sentence deltas, flat index f = b*(S-1)+s.
  float dSum = 0.f, pmSum = 0.f;
  for (int f = tid; f < NPAIR; f += 256) {
    const int b = f / (S_ - 1), s = f % (S_ - 1);
    const int l0 = labels[b * S_ + s];
    const int l1 = labels[b * S_ + s + 1];
    const float pm = ((l0 != 0 && l0 != 1) && (l1 != 0 && l1 != 1)) ? 1.f : 0.f;
    const int r0 = s * B_ + b, r1 = (s + 1) * B_ + b;
    const float* st0 = stats + (size_t)r0 * 8;
    const float* st1 = stats + (size_t)r1 * 8;
    const float nx0 = fmaxf(sqrtf(st0[0]), 1e-6f), nx1 = fmaxf(sqrtf(st1[0]), 1e-6f);
    const float ng0 = fmaxf(sqrtf(st0[1]), 1e-6f), ng1 = fmaxf(sqrtf(st1[1]), 1e-6f);
    const float dOut = st0[3] / (nx0 * nx1);
    const float dTgt = st0[4] / (ng0 * ng1);
    s_do[f] = dOut; s_dt[f] = dTgt; s_pm[f] = pm;
    const float dd = dOut - dTgt;
    dSum  += pm * dd * dd;
    pmSum += pm;
  }
  __syncthreads();

  vmSum   = blockReduceSum(vmSum, s_red);
  sqSum   = blockReduceSum(sqSum, s_red);
  cosSum  = blockReduceSum(cosSum, s_red);
  ceSum   = blockReduceSum(ceSum, s_red);
  keepSum = blockReduceSum(keepSum, s_red);
  dSum    = blockReduceSum(dSum, s_red);
  pmSum   = blockReduceSum(pmSum, s_red);

  if (tid == 0) {
    // Delta-of-delta: stable partition puts valid deltas first in flat order;
    // only consecutive packed valid pairs (n < K-1) survive the mask.
    float acc = 0.f, po = 0.f, pt = 0.f;
    int cnt = 0, have = 0;
    for (int f = 0; f < NPAIR; ++f) {
      if (s_pm[f] != 0.f) {
        const float co = s_do[f], ct = s_dt[f];
        if (have) {
          const float deno = (po != 0.f) ? po : 1e-6f;
          const float dent = (pt != 0.f) ? pt : 1e-6f;
          const float ddo = (co - po) / deno / 100.f;
          const float ddt = (ct - pt) / dent / 100.f;
          const float df = ddo - ddt;
          acc += df * df;
          ++cnt;
        }
        po = co; pt = ct; have = 1;
      }
    }
    const float loss_dd    = acc / fmaxf((float)cnt, 1.f);
    const float loss_mse   = sqrtf(sqSum / fmaxf(vmSum * (float)D_, 1.f) + 1e-6f);
    const float loss_cos   = cosSum / fmaxf(vmSum, 1.f);
    const float loss_delta = dSum / fmaxf(pmSum, 1.f);
    const float loss_ce    = ceSum / fmaxf(keepSum, 1.f);
    out[0] = loss_mse;
    out[1] = loss_cos;
    out[2] = loss_delta;
    out[3] = loss_dd;
    out[4] = loss_ce;
    out[5] = loss_mse + loss_cos + loss_delta + loss_dd + loss_ce;
  }
}

// ---------------------------------------------------------------------------
extern "C" void kernel_launch(void* const* d_in, const int* in_sizes, int n_in,
                              void* d_out, int out_size, void* d_ws, size_t ws_size,
                              hipStream_t stream) {
  const float* X      = (const float*)d_in[0];  // outputs_senemb (S,B,D)
  const float* logits = (const float*)d_in[1];  // (B,S,V)
  const float* G      = (const float*)d_in[2];  // labels_senembeddings (S,B,D)
  const int*   labels = (const int*)d_in[3];    // (B,S) int32
  float* out = (float*)d_out;

  float* stats = (float*)d_ws;          // NROWS * 8 floats (64 KB)
  float* nll   = stats + NROWS * 8;     // NROWS floats (8 KB)

  gram_kernel <<< (S_ / 16) * B_, 32, 0, stream >>>(X, G, stats);
  ce_kernel   <<< NROWS, 256, 0, stream >>>(logits, labels, nll);
  final_kernel<<< 1, 256, 0, stream >>>(stats, nll, labels, out);
}